// GNNNetwork_20469814133495
// MI455X (gfx1250) — compile-verified
//
#include <hip/hip_runtime.h>
#include <hip/hip_bf16.h>
#include <cstdint>

#define GN 50000
#define GD 64
#define GH 4
#define GE 800000
#define GL 3

#define WSTR 72   // LDS stride for 64x64 weight tile: 2*WSTR % 64 == 16 -> halves hit disjoint banks
#define CSTR 66   // LDS stride for 32x64 C tile: 8*CSTR % 64 == 16 -> halves hit disjoint banks

typedef __attribute__((ext_vector_type(2))) float v2f;
typedef __attribute__((ext_vector_type(8))) float v8f;

// ---------------- float atomic max via integer ordering trick ----------------
__device__ __forceinline__ void atomicMaxF(float* addr, float val) {
    if (val >= 0.0f) {
        atomicMax((int*)addr, __float_as_int(val));
    } else {
        atomicMin((unsigned int*)addr, __float_as_uint(val));
    }
}

// cooperative 64x64 weight load into padded LDS (vectorized float4)
__device__ __forceinline__ void stage_w(float* sW, const float* __restrict__ W) {
    for (int i = threadIdx.x; i < 1024; i += 256) {
        const float4 t = ((const float4*)W)[i];
        const int r = (i * 4) >> 6, c = (i * 4) & 63;
        *(float4*)&sW[r * WSTR + c] = t;
    }
}

// ---------------- fused Q/K/V projection ----------------
// One wave: 16x16 tile of q,k,v. A fragments loaded once, 3 WMMA chains (48 wmma).
__global__ __launch_bounds__(256) void gemm_qkv(
    const float* __restrict__ X,
    const float* __restrict__ Wq, const float* __restrict__ Wk, const float* __restrict__ Wv,
    const float* __restrict__ bq, const float* __restrict__ bk, const float* __restrict__ bv,
    float* __restrict__ q, float* __restrict__ k, float* __restrict__ v, int n)
{
    __shared__ float sWq[64 * WSTR], sWk[64 * WSTR], sWv[64 * WSTR];
    stage_w(sWq, Wq); stage_w(sWk, Wk); stage_w(sWv, Wv);
    __syncthreads();

    const int lane = threadIdx.x & 31;
    const int wave = threadIdx.x >> 5;
    const int row0 = blockIdx.x * 32 + (wave >> 2) * 16;
    const int col0 = (wave & 3) * 16;
    if (row0 >= n) return;                  // wave-uniform

    const int half = lane >> 4;
    const int l16  = lane & 15;
    const int mA   = row0 + l16;
    const int koff = half * 2;
    const int colB = col0 + l16;
    const float* __restrict__ Arow = X + (size_t)mA * GD;

    v8f cq = {}, ck = {}, cv = {};
#pragma unroll
    for (int k0 = 0; k0 < GD; k0 += 4) {
        const int ka = k0 + koff;
        v2f a, b;
        a.x = Arow[ka]; a.y = Arow[ka + 1];
        b.x = sWq[ka * WSTR + colB]; b.y = sWq[(ka + 1) * WSTR + colB];
        cq = __builtin_amdgcn_wmma_f32_16x16x4_f32(false, a, false, b, (short)0, cq, false, false);
        b.x = sWk[ka * WSTR + colB]; b.y = sWk[(ka + 1) * WSTR + colB];
        ck = __builtin_amdgcn_wmma_f32_16x16x4_f32(false, a, false, b, (short)0, ck, false, false);
        b.x = sWv[ka * WSTR + colB]; b.y = sWv[(ka + 1) * WSTR + colB];
        cv = __builtin_amdgcn_wmma_f32_16x16x4_f32(false, a, false, b, (short)0, cv, false, false);
    }

    const float bqc = bq[colB], bkc = bk[colB], bvc = bv[colB];
#pragma unroll
    for (int r = 0; r < 8; ++r) {
        const size_t m = (size_t)(row0 + r + half * 8) * GD + colB;
        q[m] = cq[r] + bqc;
        k[m] = ck[r] + bkc;
        v[m] = cv[r] + bvc;
    }
}

// ---------------- GEMM + bias + residual + LayerNorm, fully fused ----------------
// Block covers 32 complete rows -> LN stats are block-local via LDS C tile.
template<bool HAS_DENOM>
__global__ __launch_bounds__(256) void gemm_ln(
    const float* __restrict__ A, const float* __restrict__ W,
    const float* __restrict__ bias, const float* __restrict__ resid,
    const float* __restrict__ denom,
    const float* __restrict__ g, const float* __restrict__ be,
    float* __restrict__ out, int n)
{
    __shared__ float sW[64 * WSTR];
    __shared__ float sC[32 * CSTR];
    stage_w(sW, W);
    __syncthreads();

    const int lane = threadIdx.x & 31;
    const int wave = threadIdx.x >> 5;
    const int lr0  = (wave >> 2) * 16;
    const int row0 = blockIdx.x * 32 + lr0;
    const int col0 = (wave & 3) * 16;
    const int half = lane >> 4;
    const int l16  = lane & 15;

    if (row0 < n) {                          // wave-uniform
        const int mA   = row0 + l16;
        const int koff = half * 2;
        const int colB = col0 + l16;
        const float* __restrict__ Arow = A + (size_t)mA * GD;

        float dinv0 = 1.0f, dinv1 = 1.0f, dinv2 = 1.0f, dinv3 = 1.0f;
        if constexpr (HAS_DENOM) {
            const float* dr = denom + (size_t)mA * GH;
            dinv0 = 1.0f / dr[0]; dinv1 = 1.0f / dr[1];
            dinv2 = 1.0f / dr[2]; dinv3 = 1.0f / dr[3];
        }

        v8f c = {};
#pragma unroll
        for (int k0 = 0; k0 < GD; k0 += 4) {
            const int ka = k0 + koff;
            v2f a, b;
            a.x = Arow[ka]; a.y = Arow[ka + 1];
            if constexpr (HAS_DENOM) {
                const int hh = ka >> 4;
                const float s = (hh == 0) ? dinv0 : (hh == 1) ? dinv1
                              : (hh == 2) ? dinv2 : dinv3;
                a.x *= s; a.y *= s;
            }
            b.x = sW[ka * WSTR + colB];
            b.y = sW[(ka + 1) * WSTR + colB];
            c = __builtin_amdgcn_wmma_f32_16x16x4_f32(false, a, false, b, (short)0, c, false, false);
        }

        const float bcol = bias[colB];
#pragma unroll
        for (int r = 0; r < 8; ++r) {
            const int lr = lr0 + r + half * 8;
            const size_t m = (size_t)(row0 + r + half * 8) * GD + colB;
            sC[lr * CSTR + colB] = c[r] + bcol + resid[m];
        }
    }
    __syncthreads();

    // LayerNorm: 8 waves x 4 rows each over the 32-row LDS tile
#pragma unroll
    for (int rr = 0; rr < 4; ++rr) {
        const int lr  = wave * 4 + rr;
        const int row = blockIdx.x * 32 + lr;
        if (row < n) {
            const float x0 = sC[lr * CSTR + lane];
            const float x1 = sC[lr * CSTR + lane + 32];
            float s = x0 + x1;
#pragma unroll
            for (int m = 16; m >= 1; m >>= 1) s += __shfl_xor(s, m, 32);
            const float mean = s * (1.0f / 64.0f);
            const float d0 = x0 - mean, d1 = x1 - mean;
            float vs = d0 * d0 + d1 * d1;
#pragma unroll
            for (int m = 16; m >= 1; m >>= 1) vs += __shfl_xor(vs, m, 32);
            const float inv = rsqrtf(vs * (1.0f / 64.0f) + 1e-5f);
            out[(size_t)row * GD + lane]      = d0 * inv * g[lane]      + be[lane];
            out[(size_t)row * GD + lane + 32] = d1 * inv * g[lane + 32] + be[lane + 32];
        }
    }
}

// ---------------- plain GEMM + bias + ReLU (MLP hidden) ----------------
__global__ __launch_bounds__(256) void gemm_relu(
    const float* __restrict__ A, const float* __restrict__ W,
    const float* __restrict__ bias, float* __restrict__ out, int n)
{
    __shared__ float sW[64 * WSTR];
    stage_w(sW, W);
    __syncthreads();

    const int lane = threadIdx.x & 31;
    const int wave = threadIdx.x >> 5;
    const int row0 = blockIdx.x * 32 + (wave >> 2) * 16;
    const int col0 = (wave & 3) * 16;
    if (row0 >= n) return;

    const int half = lane >> 4;
    const int l16  = lane & 15;
    const int mA   = row0 + l16;
    const int koff = half * 2;
    const int colB = col0 + l16;
    const float* __restrict__ Arow = A + (size_t)mA * GD;

    v8f c = {};
#pragma unroll
    for (int k0 = 0; k0 < GD; k0 += 4) {
        const int ka = k0 + koff;
        v2f a, b;
        a.x = Arow[ka]; a.y = Arow[ka + 1];
        b.x = sW[ka * WSTR + colB];
        b.y = sW[(ka + 1) * WSTR + colB];
        c = __builtin_amdgcn_wmma_f32_16x16x4_f32(false, a, false, b, (short)0, c, false, false);
    }

    const float bcol = bias[colB];
#pragma unroll
    for (int r = 0; r < 8; ++r) {
        const size_t m = (size_t)(row0 + r + half * 8) * GD + colB;
        out[m] = fmaxf(c[r] + bcol, 0.0f);
    }
}

// ---------------- attention scratch init ----------------
__global__ __launch_bounds__(256) void att_init(
    float* __restrict__ smax, float* __restrict__ denom,
    float* __restrict__ agg, int n)
{
    const int idx = blockIdx.x * blockDim.x + threadIdx.x;
    if (idx < n * GH) { smax[idx] = -INFINITY; denom[idx] = 0.0f; }
    if (idx < n * GD) { agg[idx] = 0.0f; }
}

// ---------------- per-(edge,head) score + segment max ----------------
__global__ __launch_bounds__(256) void edge_scores(
    const float* __restrict__ q, const float* __restrict__ k,
    const long long* __restrict__ ei, const float* __restrict__ ea,
    float* __restrict__ alpha, float* __restrict__ smax)
{
    const int idx = blockIdx.x * blockDim.x + threadIdx.x;
    if (idx >= GE * GH) return;
    const int e = idx >> 2, h = idx & 3;
    const int src = (int)ei[e];           // row: source
    const int dst = (int)ei[GE + e];      // col: destination
    const float4* qp = (const float4*)(q + (size_t)dst * GD + h * 16);
    const float4* kp = (const float4*)(k + (size_t)src * GD + h * 16);
    float s = 0.0f;
#pragma unroll
    for (int i = 0; i < 4; ++i) {
        const float4 a = qp[i], b = kp[i];
        s += a.x * b.x + a.y * b.y + a.z * b.z + a.w * b.w;
    }
    s *= 0.25f * ea[e];                   // inv_sqrt(HD)=0.25 and edge_attr
    alpha[idx] = s;
    atomicMaxF(&smax[(size_t)dst * GH + h], s);
}

// ---------------- alpha = exp(s - smax[dst]); segment sum ----------------
__global__ __launch_bounds__(256) void edge_alpha(
    const long long* __restrict__ ei, float* __restrict__ alpha,
    const float* __restrict__ smax, float* __restrict__ denom)
{
    const int idx = blockIdx.x * blockDim.x + threadIdx.x;
    if (idx >= GE * GH) return;
    const int e = idx >> 2, h = idx & 3;
    const int dst = (int)ei[GE + e];
    const float a = expf(alpha[idx] - smax[(size_t)dst * GH + h]);
    alpha[idx] = a;
    atomicAdd(&denom[(size_t)dst * GH + h], a);
}

// ---------------- agg[dst] += alpha * v[src] (unnormalized) ----------------
__global__ __launch_bounds__(256) void edge_agg(
    const long long* __restrict__ ei, const float* __restrict__ alpha,
    const float* __restrict__ v, float* __restrict__ agg)
{
    const int idx = blockIdx.x * blockDim.x + threadIdx.x;
    if (idx >= GE * GH) return;
    const int e = idx >> 2, h = idx & 3;
    const int src = (int)ei[e];
    const int dst = (int)ei[GE + e];
    const float a = alpha[idx];
    const float* vp = v + (size_t)src * GD + h * 16;
    float* ap = agg + (size_t)dst * GD + h * 16;
#pragma unroll
    for (int d = 0; d < 16; ++d) atomicAdd(&ap[d], a * vp[d]);
}

extern "C" void kernel_launch(void* const* d_in, const int* in_sizes, int n_in,
                              void* d_out, int out_size, void* d_ws, size_t ws_size,
                              hipStream_t stream) {
    const float*     feats = (const float*)d_in[0];
    const long long* ei    = (const long long*)d_in[1];
    const float*     ea    = (const float*)d_in[2];
    const float* Wq = (const float*)d_in[3];
    const float* Wk = (const float*)d_in[4];
    const float* Wv = (const float*)d_in[5];
    const float* Wo = (const float*)d_in[6];
    const float* W1 = (const float*)d_in[7];
    const float* W2 = (const float*)d_in[8];
    const float* bq = (const float*)d_in[9];
    const float* bk = (const float*)d_in[10];
    const float* bv = (const float*)d_in[11];
    const float* bo = (const float*)d_in[12];
    const float* b1 = (const float*)d_in[13];
    const float* b2 = (const float*)d_in[14];
    const float* g1 = (const float*)d_in[15];
    const float* be1= (const float*)d_in[16];
    const float* g2 = (const float*)d_in[17];
    const float* be2= (const float*)d_in[18];
    float* out = (float*)d_out;

    // workspace layout (floats): q,k,v,agg + alpha + smax + denom  (~64 MB)
    float* ws = (float*)d_ws;
    const size_t ND = (size_t)GN * GD;
    float* q     = ws;                        // N*64  (also reused as X1 after LN1)
    float* kbuf  = q    + ND;                 // N*64  (also reused as MLP hidden)
    float* v     = kbuf + ND;                 // N*64
    float* agg   = v    + ND;                 // N*64
    float* alpha = agg  + ND;                 // E*H
    float* smax  = alpha + (size_t)GE * GH;   // N*H
    float* denom = smax  + (size_t)GN * GH;   // N*H
    (void)ws_size; (void)in_sizes; (void)n_in; (void)out_size;

    const dim3 gemm_grid((GN + 31) / 32), blk(256);
    const dim3 eh_grid((GE * GH + 255) / 256);
    const dim3 ini_grid((GN * GD + 255) / 256);

    for (int l = 0; l < GL; ++l) {
        const float* X = (l == 0) ? feats : out + (size_t)(l - 1) * ND;
        const size_t wo  = (size_t)l * GD * GD;   // weight offset
        const size_t bo_ = (size_t)l * GD;        // bias/gain offset

        // fused Q/K/V projection (X read once)
        gemm_qkv<<<gemm_grid, blk, 0, stream>>>(X, Wq + wo, Wk + wo, Wv + wo,
                                                bq + bo_, bk + bo_, bv + bo_,
                                                q, kbuf, v, GN);

        // edge-indexed scatter-softmax attention
        att_init   <<<ini_grid, blk, 0, stream>>>(smax, denom, agg, GN);
        edge_scores<<<eh_grid, blk, 0, stream>>>(q, kbuf, ei, ea, alpha, smax);
        edge_alpha <<<eh_grid, blk, 0, stream>>>(ei, alpha, smax, denom);
        edge_agg   <<<eh_grid, blk, 0, stream>>>(ei, alpha, v, agg);

        // att = (agg/denom)@Wo + bo + X, LayerNorm1 fused  -> q (reused as X1)
        gemm_ln<true><<<gemm_grid, blk, 0, stream>>>(agg, Wo + wo, bo + bo_, X, denom,
                                                     g1 + bo_, be1 + bo_, q, GN);

        // hidden = relu(X1@W1 + b1) -> kbuf
        gemm_relu<<<gemm_grid, blk, 0, stream>>>(q, W1 + wo, b1 + bo_, kbuf, GN);

        // out_l = LN2(hidden@W2 + b2 + X1) -> layer output (feeds next layer)
        gemm_ln<false><<<gemm_grid, blk, 0, stream>>>(kbuf, W2 + wo, b2 + bo_, q, nullptr,
                                                      g2 + bo_, be2 + bo_,
                                                      out + (size_t)l * ND, GN);
    }
}